// Quantizer_84559316124504
// MI455X (gfx1250) — compile-verified
//
#include <hip/hip_runtime.h>

// Shapes (fixed by the reference): B=32,H=32,W=32,D=64,K=512, N=B*H*W=32768
#define NROWS 32768
#define DDIM  64
#define KDIM  512
#define AP    65   // padded pitch for 64-wide LDS tiles (bank-conflict free)
#define LP    516  // padded pitch for 512-wide logit tile (516%64==4)

typedef __attribute__((ext_vector_type(2))) float v2f;
typedef __attribute__((ext_vector_type(8))) float v8f;

__global__ void vq_zero_loss(float* loss) {
    if (threadIdx.x == 0) *loss = 0.0f;
}

__global__ __launch_bounds__(32)
void vq_main(const float* __restrict__ mu,   // [B,H,W,D] = [N,D]
             const float* __restrict__ var,  // [B,D,H,W]
             const float* __restrict__ E,    // [K,D]
             const float* __restrict__ gum,  // [N,K]
             const float* __restrict__ temp, // [1]
             float* __restrict__ out_q,      // [B,D,H,W]
             float* __restrict__ out_idx,    // [N] (as float)
             float* __restrict__ loss)       // [1]
{
    __shared__ float A1s[16 * AP];  // 2*w*mu tile
    __shared__ float Ws [16 * AP];  // w tile
    __shared__ float MUs[16 * AP];  // mu tile
    __shared__ float Ets[16 * AP];  // staged E rows for current column tile
    __shared__ float Ls [16 * LP];  // logits, later overwritten with encodings
    __shared__ float Qs [16 * AP];  // quant tile

    const int lane = threadIdx.x;       // wave32: lane id
    const int n0   = blockIdx.x * 16;   // first row of this M-tile
    const float invT = 1.0f / temp[0];

    // ---- load mu / var tiles; var is [B,D,H,W]: n = b*1024 + hw ----
    for (int i = lane; i < 16 * 64; i += 32) {
        int r = i >> 6, d = i & 63;
        int n = n0 + r;
        int b = n >> 10, hw = n & 1023;
        float m = mu[(size_t)n * 64 + d];
        float v = var[((size_t)(b * 64 + d)) * 1024 + hw];
        v = (v < 1e-10f) ? 1e-10f : v;
        float w = 0.5f / v;
        MUs[r * AP + d] = m;
        Ws [r * AP + d] = w;
        A1s[r * AP + d] = 2.0f * w * m;
    }
    __syncthreads();

    const int half2 = (lane >= 16) ? 2 : 0;  // A/B fragment K sub-offset
    const int c     = lane & 15;             // M row (A) / N col (B) within tile
    const int roff  = (lane >= 16) ? 8 : 0;  // C/D fragment M offset

    // ---- hoist A fragments (loop-invariant across all 32 column tiles) ----
    v2f a1f[16], a2f[16];
    #pragma unroll
    for (int kk4 = 0; kk4 < 16; ++kk4) {
        a1f[kk4].x = A1s[c * AP + kk4 * 4 + half2];
        a1f[kk4].y = A1s[c * AP + kk4 * 4 + half2 + 1];
        a2f[kk4].x = Ws [c * AP + kk4 * 4 + half2];
        a2f[kk4].y = Ws [c * AP + kk4 * 4 + half2 + 1];
    }

    // ---- logit GEMM: logit = (2*w*mu) @ E^T  -  w @ (E*E)^T  (shift dropped) ----
    for (int t = 0; t < 32; ++t) {
        const int k0 = t * 16;
        for (int i = lane; i < 16 * 64; i += 32) {
            int r = i >> 6, d = i & 63;
            Ets[r * AP + d] = E[(size_t)(k0 + r) * 64 + d];
        }
        __syncthreads();

        v8f acc = {0.f, 0.f, 0.f, 0.f, 0.f, 0.f, 0.f, 0.f};
        #pragma unroll
        for (int kk4 = 0; kk4 < 16; ++kk4) {
            float e0 = Ets[c * AP + kk4 * 4 + half2];
            float e1 = Ets[c * AP + kk4 * 4 + half2 + 1];
            v2f b1, b2;
            b1.x = e0;       b1.y = e1;
            b2.x = -e0 * e0; b2.y = -e1 * e1;
            acc = __builtin_amdgcn_wmma_f32_16x16x4_f32(false, a1f[kk4], false, b1,
                                                        (short)0, acc, false, false);
            acc = __builtin_amdgcn_wmma_f32_16x16x4_f32(false, a2f[kk4], false, b2,
                                                        (short)0, acc, false, false);
        }
        #pragma unroll
        for (int r = 0; r < 8; ++r)
            Ls[(r + roff) * LP + k0 + c] = acc[r];
        __syncthreads();
    }

    // ---- per-row softmax stats, argmax, gumbel-softmax encodings ----
    float kd_acc = 0.0f;  // sum p*log p (partial)
    for (int r = 0; r < 16; ++r) {
        // prefetch next gumbel row (2 KB): 32 lanes x 64B lines
        if (r + 1 < 16)
            __builtin_prefetch(&gum[(size_t)(n0 + r + 1) * 512 + lane * 16], 0, 0);

        float* Lrow = &Ls[r * LP];
        float lv[16];
        float bv = -3.4e38f; int bi = 0x7fffffff;
        #pragma unroll
        for (int j = 0; j < 16; ++j) {
            float l = Lrow[lane + j * 32];
            lv[j] = l;
            int idx = lane + j * 32;
            if (l > bv || (l == bv && idx < bi)) { bv = l; bi = idx; }
        }
        float mx = bv;
        for (int off = 16; off; off >>= 1)
            mx = fmaxf(mx, __shfl_xor(mx, off, 32));
        for (int off = 16; off; off >>= 1) {
            float ov = __shfl_xor(bv, off, 32);
            int   oi = __shfl_xor(bi, off, 32);
            if (ov > bv || (ov == bv && oi < bi)) { bv = ov; bi = oi; }
        }
        float s = 0.0f;
        float ev[16];
        #pragma unroll
        for (int j = 0; j < 16; ++j) { ev[j] = __expf(lv[j] - mx); s += ev[j]; }
        for (int off = 16; off; off >>= 1) s += __shfl_xor(s, off, 32);
        float logS = __logf(s);
        float invS = 1.0f / s;
        #pragma unroll
        for (int j = 0; j < 16; ++j) {
            float p  = ev[j] * invS;
            float lp = (lv[j] - mx) - logS;
            kd_acc += p * lp;
        }
        // gumbel-softmax encodings overwrite the logit row
        const float* grow = &gum[(size_t)(n0 + r) * 512];
        float zx = -3.4e38f;
        float zv[16];
        #pragma unroll
        for (int j = 0; j < 16; ++j) {
            float z = (lv[j] + grow[lane + j * 32]) * invT;
            zv[j] = z;
            zx = fmaxf(zx, z);
        }
        for (int off = 16; off; off >>= 1)
            zx = fmaxf(zx, __shfl_xor(zx, off, 32));
        float sz = 0.0f;
        #pragma unroll
        for (int j = 0; j < 16; ++j) { zv[j] = __expf(zv[j] - zx); sz += zv[j]; }
        for (int off = 16; off; off >>= 1) sz += __shfl_xor(sz, off, 32);
        float invSz = 1.0f / sz;
        #pragma unroll
        for (int j = 0; j < 16; ++j)
            Lrow[lane + j * 32] = zv[j] * invSz;
        if (lane == 0) out_idx[n0 + r] = (float)bi;
    }
    __syncthreads();

    // ---- quant GEMM: quant[16,64] = enc[16,512] @ E[512,64] ----
    for (int t = 0; t < 4; ++t) {
        const int d0 = t * 16;
        v8f acc = {0.f, 0.f, 0.f, 0.f, 0.f, 0.f, 0.f, 0.f};
        #pragma unroll 8
        for (int kk = 0; kk < 512; kk += 4) {
            v2f a, b;
            a.x = Ls[c * LP + kk + half2];
            a.y = Ls[c * LP + kk + half2 + 1];
            b.x = E[(size_t)(kk + half2) * 64 + d0 + c];
            b.y = E[(size_t)(kk + half2 + 1) * 64 + d0 + c];
            acc = __builtin_amdgcn_wmma_f32_16x16x4_f32(false, a, false, b,
                                                        (short)0, acc, false, false);
        }
        #pragma unroll
        for (int r = 0; r < 8; ++r)
            Qs[(r + roff) * AP + d0 + c] = acc[r];
    }
    __syncthreads();

    // ---- kld_continuous partial + loss accumulation ----
    float kc = 0.0f;
    if (lane < 16) {
        #pragma unroll
        for (int d = 0; d < 64; ++d) {
            float diff = Qs[lane * AP + d] - MUs[lane * AP + d];
            kc += diff * diff * Ws[lane * AP + d];
        }
    }
    float tot = kd_acc * (1.0f / 32.0f) + kc * (1.0f / 32768.0f);
    for (int off = 16; off; off >>= 1) tot += __shfl_xor(tot, off, 32);
    if (lane == 0) atomicAdd(loss, tot);

    // ---- quantized output, transposed to [B,D,H,W] ----
    for (int i = lane; i < 16 * 64; i += 32) {
        int r = i & 15, d = i >> 4;
        int n = n0 + r;
        int b = n >> 10, hw = n & 1023;
        out_q[((size_t)(b * 64 + d)) * 1024 + hw] = Qs[r * AP + d];
    }
}

extern "C" void kernel_launch(void* const* d_in, const int* in_sizes, int n_in,
                              void* d_out, int out_size, void* d_ws, size_t ws_size,
                              hipStream_t stream) {
    const float* mu   = (const float*)d_in[0];
    const float* var  = (const float*)d_in[1];
    const float* E    = (const float*)d_in[2];
    const float* gum  = (const float*)d_in[3];
    const float* temp = (const float*)d_in[4];

    float* out      = (float*)d_out;
    float* out_q    = out;                          // 32768*64 floats
    float* out_idx  = out + (size_t)NROWS * DDIM;   // 32768 floats
    float* loss     = out_idx + NROWS;              // 1 float

    vq_zero_loss<<<1, 32, 0, stream>>>(loss);
    vq_main<<<NROWS / 16, 32, 0, stream>>>(mu, var, E, gum, temp,
                                           out_q, out_idx, loss);
}